// CSIEncoder_38749194945148
// MI455X (gfx1250) — compile-verified
//
#include <hip/hip_runtime.h>

typedef float v2f __attribute__((ext_vector_type(2)));
typedef float v4f __attribute__((ext_vector_type(4)));
typedef float v8f __attribute__((ext_vector_type(8)));

#define LN_EPS   1e-5f
#define NGROUPS  32768   // (4*64*4*512) tokens / 16 tokens per group
#define NBLOCKS  512     // fixed: wave stride = 4096 groups (mult. of 128),
                         // so f0 and u are wave-invariant across the loop

#if __has_builtin(__builtin_amdgcn_rsqf)
#define RSQ(x) __builtin_amdgcn_rsqf(x)
#else
#define RSQ(x) rsqrtf(x)
#endif

// Gc/P linear index for symmetric 6x6, i<=j
#define GC(i, j) Gc[(i)*6 - (i)*((i) + 1) / 2 + (j)]

// One wave32 per 16-token group. Pre-LN x = A(16x4) * B(4x256) via 16x
// V_WMMA_F32_16X16X4_F32 (K = re, im, 1, m), ln_gamma pre-folded into B.
// Column permutation d = 16*n + t gives each lane 16 consecutive d across
// the 16 tiles -> outputs packed 4-wide into global_store_b128 (32/group).
// LayerNorm stats come from per-wave weight means + Gram matrix (exact
// finite-sum identities), so tiles stream out with 32 acc VGPRs live.
__global__ __launch_bounds__(256) void csi_ln_wmma(
    const float* __restrict__ csi_re, const float* __restrict__ csi_im,
    const float* __restrict__ w_c,  const float* __restrict__ b_c,
    const float* __restrict__ w_s,  const float* __restrict__ b_s,
    const float* __restrict__ w_u,  const float* __restrict__ b_u,
    const float* __restrict__ w_f,  const float* __restrict__ b_f,
    const float* __restrict__ ln_g, const float* __restrict__ ln_b,
    float* __restrict__ out)
{
  const int  lane = threadIdx.x & 31;
  const int  col  = lane & 15;      // tile column n; also token m for A/stats
  const bool hi   = lane >= 16;     // hi half: K=2,3 of A / rows K2,K3 of B

  const int wavesPerBlk = blockDim.x >> 5;
  const int gwave = blockIdx.x * wavesPerBlk + (threadIdx.x >> 5);
  const int nwave = gridDim.x * wavesPerBlk;   // 4096

  // Wave-invariant token-index features (stride 4096 preserves them)
  const float ff0 = (float)((gwave & 31) << 4);   // base subcarrier f0
  const float fu  = (float)((gwave >> 5) & 3);    // UE antenna index

  // ---- Per-wave setup: weight columns d = 16*col + t (consecutive per
  // lane -> b128 loads). Vector ids: 0=wc0 1=wc1 2=bsum 3=ws 4=wu 5=wf
  float W[6][16], g[16], bb[16];
#pragma unroll
  for (int t = 0; t < 16; ++t) {
    const int d = (col << 4) + t;
    W[0][t] = w_c[2 * d + 0];
    W[1][t] = w_c[2 * d + 1];
    W[2][t] = b_c[d] + b_s[d] + b_u[d] + b_f[d];
    W[3][t] = w_s[d];
    W[4][t] = w_u[d];
    W[5][t] = w_f[d];
    g[t]  = ln_g[d];
    bb[t] = ln_b[d];
  }

  // Sums + raw dot products (27 scalars) over the 16 d per lane, then
  // across the 16 lanes of each half (halves identical -> wave-uniform).
  float S[6], P[21];
#pragma unroll
  for (int i = 0; i < 6; ++i) S[i] = 0.0f;
#pragma unroll
  for (int k = 0; k < 21; ++k) P[k] = 0.0f;
#pragma unroll
  for (int t = 0; t < 16; ++t) {
#pragma unroll
    for (int i = 0; i < 6; ++i) S[i] += W[i][t];
    int k = 0;
#pragma unroll
    for (int i = 0; i < 6; ++i)
#pragma unroll
      for (int j = i; j < 6; ++j) {
        P[k] = fmaf(W[i][t], W[j][t], P[k]);
        ++k;
      }
  }
#pragma unroll
  for (int m = 1; m <= 8; m <<= 1) {
#pragma unroll
    for (int i = 0; i < 6; ++i) S[i] += __shfl_xor(S[i], m, 32);
#pragma unroll
    for (int k = 0; k < 21; ++k) P[k] += __shfl_xor(P[k], m, 32);
  }

  float mean[6];
#pragma unroll
  for (int i = 0; i < 6; ++i) mean[i] = S[i] * (1.0f / 256.0f);

  float Gc[21];   // centered Gram: dot/256 - mean_i*mean_j
  {
    int k = 0;
#pragma unroll
    for (int i = 0; i < 6; ++i)
#pragma unroll
      for (int j = i; j < 6; ++j) {
        Gc[k] = fmaf(-mean[i], mean[j], P[k] * (1.0f / 256.0f));
        ++k;
      }
  }

  // Fold wave-invariant (fu, ff0) into fs-polynomial coefficients.
  const float m0 = mean[0], m1 = mean[1], m3 = mean[3], m5 = mean[5];
  const float mgb0 = mean[2] + fu * mean[4] + ff0 * mean[5];
  const float g0g0 = GC(0,2) + fu * GC(0,4) + ff0 * GC(0,5), G03 = GC(0,3);
  const float g1g0 = GC(1,2) + fu * GC(1,4) + ff0 * GC(1,5), G13 = GC(1,3);
  const float gfg0 = GC(2,5) + fu * GC(4,5) + ff0 * GC(5,5), G35 = GC(3,5);
  const float ggg0 = GC(2,2) + fu * fu * GC(4,4) + ff0 * ff0 * GC(5,5)
                   + 2.0f * (fu * GC(2,4) + ff0 * GC(2,5) + fu * ff0 * GC(4,5));
  const float ggg1 = 2.0f * (GC(2,3) + fu * GC(3,4) + ff0 * GC(3,5));
  const float G33 = GC(3,3);
  const float G00 = GC(0,0), G11 = GC(1,1), G55 = GC(5,5);
  const float G01 = GC(0,1), G05 = GC(0,5), G15 = GC(1,5);

  // B-row registers (ln_gamma folded; WMMA outputs x*g):
  //  rAs : g*wc0 (lo, row K0) | g*ws (hi, fs-varying part of K2)
  //  rY  : g*wc1 (lo, row K1) | g*wf (hi, row K3)  -- group-invariant
  //  gb0h: lo = rAs (so bm.x = fma(fsh, rAs, gb0h) needs no cndmask)
  //        hi = g*(bsum + u*wu + f0*wf)  (fs-invariant part of K2)
  float rAs[16], rY[16], gb0h[16];
#pragma unroll
  for (int t = 0; t < 16; ++t) {
    rAs[t]  = g[t] * (hi ? W[3][t] : W[0][t]);
    rY[t]   = g[t] * (hi ? W[5][t] : W[1][t]);
    gb0h[t] = hi ? g[t] * (W[2][t] + fu * W[4][t] + ff0 * W[5][t]) : rAs[t];
  }

  // Per-lane 32-bit store offsets: token (r+src), columns d=16*col..+15
  const int src = hi ? 8 : 0;
  unsigned voffBase[8];
#pragma unroll
  for (int r = 0; r < 8; ++r)
    voffBase[r] = (unsigned)(((r + src) << 10) + (col << 6));
  const unsigned loffCol = (unsigned)(col << 2);

  int bidx[8];                      // byte indices for ds_bpermute
#pragma unroll
  for (int r = 0; r < 8; ++r) bidx[r] = (r + src) << 2;

  const float aHiX = 1.0f, aHiY = (float)col;   // A hi half: K2=1, K3=m

  // ---- Main loop over 16-token groups ---------------------------------
  for (int grp = gwave; grp < NGROUPS; grp += nwave) {
    const float fs = (float)((grp >> 7) & 63);       // BS antenna index
    const unsigned goff = (unsigned)grp << 14;       // byte offset of group
    const unsigned loff = ((unsigned)grp << 6) + loffCol;

    const float hre = __builtin_nontemporal_load(
        (const float*)((const char*)csi_re + loff));
    const float him = __builtin_nontemporal_load(
        (const float*)((const char*)csi_im + loff));

    v2f a;
    a.x = hi ? aHiX : hre;
    a.y = hi ? aHiY : him;

    // Group stats from fs-polynomials (wave-uniform)
    const float mgb = fmaf(fs, m3,  mgb0);
    const float G0g = fmaf(fs, G03, g0g0);
    const float G1g = fmaf(fs, G13, g1g0);
    const float Gfg = fmaf(fs, G35, gfg0);
    const float Ggg = fmaf(fs, fmaf(fs, G33, ggg1), ggg0);

    // Per-token stats in "token = col" layout (exact bilinear forms)
    const float mm    = (float)col;
    const float mu_t  = hre * m0 + him * m1 + mgb + mm * m5;
    const float var_t = hre * hre * G00 + him * him * G11
                      + mm * mm * G55 + Ggg
                      + 2.0f * (hre * him * G01
                              + hre * fmaf(mm, G05, G0g)
                              + him * fmaf(mm, G15, G1g)
                              + mm * Gfg);
    const float rstd_t = RSQ(var_t + LN_EPS);
    const float q_t    = -mu_t * rstd_t;   // out = y*rstd + (q*g + beta)

    // Redistribute to (r, half) layout: independent bpermute dests
    float rsR[8], q[8];
#pragma unroll
    for (int r = 0; r < 8; ++r) {
      rsR[r] = __int_as_float(
          __builtin_amdgcn_ds_bpermute(bidx[r], __float_as_int(rstd_t)));
      q[r]   = __int_as_float(
          __builtin_amdgcn_ds_bpermute(bidx[r], __float_as_int(q_t)));
    }

    unsigned voff[8];
#pragma unroll
    for (int r = 0; r < 8; ++r) voff[r] = goff + voffBase[r];

    const float fsh = hi ? fs : 0.0f;   // lane-masked fs for the K2 build

    // 4 chunks x 4 tiles: WMMA -> normalize (2 FMA/elem) -> b128 NT store
#pragma unroll
    for (int c = 0; c < 4; ++c) {
      v8f acc[4];
#pragma unroll
      for (int j = 0; j < 4; ++j) {
        const int t = 4 * c + j;
        v2f bm;
        bm.x = fmaf(fsh, rAs[t], gb0h[t]);   // K0 (lo) | K2 (hi)
        bm.y = rY[t];                         // K1 (lo) | K3 (hi)
        v8f cz = {};
        acc[j] = __builtin_amdgcn_wmma_f32_16x16x4_f32(
            false, a, false, bm, (short)0, cz, false, false);
      }
#pragma unroll
      for (int r = 0; r < 8; ++r) {
        v4f vv;
#pragma unroll
        for (int j = 0; j < 4; ++j) {
          const int t = 4 * c + j;
          vv[j] = fmaf(acc[j][r], rsR[r], fmaf(q[r], g[t], bb[t]));
        }
        __builtin_nontemporal_store(
            vv, (v4f*)((char*)out + (size_t)voff[r] + 16 * c));
      }
    }
  }
}

extern "C" void kernel_launch(void* const* d_in, const int* in_sizes, int n_in,
                              void* d_out, int out_size, void* d_ws, size_t ws_size,
                              hipStream_t stream) {
  const float* csi_re = (const float*)d_in[0];
  const float* csi_im = (const float*)d_in[1];
  const float* w_c    = (const float*)d_in[2];
  const float* b_c    = (const float*)d_in[3];
  const float* w_s    = (const float*)d_in[4];
  const float* b_s    = (const float*)d_in[5];
  const float* w_u    = (const float*)d_in[6];
  const float* b_u    = (const float*)d_in[7];
  const float* w_f    = (const float*)d_in[8];
  const float* b_f    = (const float*)d_in[9];
  const float* ln_g   = (const float*)d_in[10];
  const float* ln_b   = (const float*)d_in[11];
  float* out = (float*)d_out;

  // 512 blocks x 256 threads = 4096 waves; 8 groups/wave amortizes the
  // one-time Gram/weight setup and keeps f0/u wave-invariant (stride 4096).
  csi_ln_wmma<<<NBLOCKS, 256, 0, stream>>>(csi_re, csi_im, w_c, b_c, w_s, b_s,
                                           w_u, b_u, w_f, b_f, ln_g, ln_b, out);
}